// HungarianMatcher_28441273434199
// MI455X (gfx1250) — compile-verified
//
#include <hip/hip_runtime.h>
#include <hip/hip_bf16.h>

typedef __attribute__((ext_vector_type(2))) float v2f;
typedef __attribute__((ext_vector_type(8))) float v8f;

#define BQ    9600   // N = B*Q = 32*300
#define NCLS  11
#define NT    3200
#define NPAD  16     // padded class dim (WMMA K coverage = 12 >= 11)
#define TT_TILES (NT / 16)          // 200
#define TN_TILES (BQ / 16)          // 600
#define WAVES_PER_BLOCK 8

// Fast f32 division: v_rcp_f32 (~1 ulp) + one Newton-Raphson step (~0.5 ulp).
// Denominators in this kernel are strictly positive (x + 1e-6), no edge cases.
__device__ __forceinline__ float fastdiv(float x, float d) {
  float r = __builtin_amdgcn_rcpf(d);
  r = __builtin_fmaf(__builtin_fmaf(-d, r, 1.0f), r, r);
  return x * r;
}

// ---------------------------------------------------------------------------
// Kernel 1: row softmax of [BQ, 11] logits -> zero-padded [BQ, 16] probs (ws)
// ---------------------------------------------------------------------------
__global__ __launch_bounds__(256)
void softmax_pad_kernel(const float* __restrict__ logits,
                        float* __restrict__ probs) {
  int n = blockIdx.x * blockDim.x + threadIdx.x;
  if (n >= BQ) return;
  const float* row = logits + (size_t)n * NCLS;
  float m = row[0];
#pragma unroll
  for (int i = 1; i < NCLS; ++i) m = fmaxf(m, row[i]);
  float e[NCLS];
  float s = 0.f;
#pragma unroll
  for (int i = 0; i < NCLS; ++i) { e[i] = expf(row[i] - m); s += e[i]; }
  float inv = fastdiv(1.0f, s);
  float* o = probs + (size_t)n * NPAD;
#pragma unroll
  for (int i = 0; i < NCLS; ++i) o[i] = e[i] * inv;
#pragma unroll
  for (int i = NCLS; i < NPAD; ++i) o[i] = 0.f;
}

// ---------------------------------------------------------------------------
// Kernel 2: one wave32 per 16x16 output tile.
//   cost_class tile via V_WMMA_F32_16X16X4_F32 (probs x one-hot, exact in f32)
//   L1 + GIoU fused elementwise on the WMMA C/D register layout.
// ---------------------------------------------------------------------------
__global__ __launch_bounds__(256)
void cost_tile_kernel(const float* __restrict__ probs,   // [BQ, 16] padded
                      const float* __restrict__ pboxes,  // [BQ, 4] cxcywh
                      const int*   __restrict__ labels,  // [NT]
                      const float* __restrict__ tboxes,  // [NT, 4] cxcywh
                      float* __restrict__ out)           // [BQ, NT]
{
  const int lane = threadIdx.x & 31;
  const int tile = blockIdx.x * WAVES_PER_BLOCK + (threadIdx.x >> 5);
  const int tn = tile / TT_TILES;
  const int tt = tile - tn * TT_TILES;
  const int n0 = tn << 4;
  const int t0 = tt << 4;

  const int half = lane >> 4;       // 0: lanes 0-15, 1: lanes 16-31
  const int l15  = lane & 15;
  const int kq   = half << 1;       // A/B K-pair base within 4-wide K slab

  // ---- class-cost tile: D[m][c] = probs[n0+m][ label[t0+c] ] via WMMA ----
  const int t     = t0 + l15;
  const int label = labels[t];
  const float* aRow = probs + (size_t)(n0 + l15) * NPAD + kq; // 8B aligned

  v8f acc = {};
#pragma unroll
  for (int kc = 0; kc < 3; ++kc) {          // K = 0..11 covers C=11
    v2f a = *(const v2f*)(aRow + kc * 4);   // A[m][kbase], A[m][kbase+1]
    const int kbase = kc * 4 + kq;
    v2f b;
    b.x = (label == kbase)     ? 1.0f : 0.0f;   // one-hot column of target t
    b.y = (label == kbase + 1) ? 1.0f : 0.0f;
    acc = __builtin_amdgcn_wmma_f32_16x16x4_f32(
        /*neg_a=*/false, a, /*neg_b=*/false, b,
        /*c_mod=*/(short)0, acc, /*reuse_a=*/false, /*reuse_b=*/false);
  }
  // acc[r] = probs[n0 + r + half*8][label[t]]  (C/D layout: VGPR r -> M=r/r+8)

  // ---- elementwise L1 + GIoU on the same slots, fused with the store ----
  const float4 tb = ((const float4*)tboxes)[t];
  const float tx0 = tb.x - 0.5f * tb.z;
  const float ty0 = tb.y - 0.5f * tb.w;
  const float tx1 = tb.x + 0.5f * tb.z;
  const float ty1 = tb.y + 0.5f * tb.w;
  const float areaT = (tx1 - tx0) * (ty1 - ty0);

  const int mbase = half << 3;
#pragma unroll
  for (int r = 0; r < 8; ++r) {
    const int n = n0 + mbase + r;
    const float4 pb = ((const float4*)pboxes)[n];

    // L1 on cxcywh coords (matches reference cdist)
    const float l1 = fabsf(pb.x - tb.x) + fabsf(pb.y - tb.y) +
                     fabsf(pb.z - tb.z) + fabsf(pb.w - tb.w);

    const float px0 = pb.x - 0.5f * pb.z;
    const float py0 = pb.y - 0.5f * pb.w;
    const float px1 = pb.x + 0.5f * pb.z;
    const float py1 = pb.y + 0.5f * pb.w;
    const float areaP = (px1 - px0) * (py1 - py0);

    const float iw = fmaxf(fminf(px1, tx1) - fmaxf(px0, tx0), 0.0f);
    const float ih = fmaxf(fminf(py1, ty1) - fmaxf(py0, ty0), 0.0f);
    const float inter = iw * ih;
    const float uni   = areaP + areaT - inter;

    const float cw = fmaxf(fmaxf(px1, tx1) - fminf(px0, tx0), 0.0f);
    const float ch = fmaxf(fmaxf(py1, ty1) - fminf(py0, ty0), 0.0f);
    const float areaC = cw * ch;

    // 2*giou = 2*inter/(uni+eps) - 2*(areaC-uni)/(areaC+eps)  (x2 scaling exact)
    const float iou2 = fastdiv(2.0f * inter, uni + 1e-6f);
    const float enc2 = fastdiv(2.0f * (areaC - uni), areaC + 1e-6f);

    // COST_CLASS*(-prob) + COST_BBOX*l1 + COST_GIOU*(-giou)
    const float cost = 5.0f * l1 - iou2 + enc2 - acc[r];
    __builtin_nontemporal_store(cost, &out[(size_t)n * NT + t]);
  }
}

// ---------------------------------------------------------------------------
extern "C" void kernel_launch(void* const* d_in, const int* in_sizes, int n_in,
                              void* d_out, int out_size, void* d_ws, size_t ws_size,
                              hipStream_t stream) {
  (void)in_sizes; (void)n_in; (void)out_size; (void)ws_size;
  const float* logits = (const float*)d_in[0];   // [32,300,11]
  const float* pboxes = (const float*)d_in[1];   // [32,300,4]
  const int*   labels = (const int*)d_in[2];     // [3200]
  const float* tboxes = (const float*)d_in[3];   // [3200,4]
  float* out   = (float*)d_out;                  // [32,300,3200]
  float* probs = (float*)d_ws;                   // needs BQ*16*4 = 614400 B

  softmax_pad_kernel<<<(BQ + 255) / 256, 256, 0, stream>>>(logits, probs);

  const int tiles  = TN_TILES * TT_TILES;              // 120000
  const int blocks = tiles / WAVES_PER_BLOCK;          // 15000
  cost_tile_kernel<<<blocks, 256, 0, stream>>>(probs, pboxes, labels, tboxes, out);
}